// RelationalDistMult_76347338653996
// MI455X (gfx1250) — compile-verified
//
#include <hip/hip_runtime.h>
#include <hip/hip_bf16.h>

typedef __attribute__((ext_vector_type(2))) float v2f;
typedef __attribute__((ext_vector_type(8))) float v8f;

#define B_   8
#define N_   1024
#define DE_  128
#define R_   8
#define DR_  64
#define D_   192

#define TILE_M 64
#define TILE_N 128
#define LDA    132   // padded LDS row stride in floats (528B: 16B aligned, spreads banks)

// ---------------------------------------------------------------------------
// Pre-kernel: c[b,r] = sum_d w[r, 128+d] * rel[b,r,d]^2   (64 scalars)
// ---------------------------------------------------------------------------
__global__ void relconst_kernel(const float* __restrict__ rel,
                                const float* __restrict__ w,
                                float* __restrict__ c) {
    int id = threadIdx.x;                 // 0..63 == b*8 + r
    if (id < B_ * R_) {
        int r = id & 7;
        const float* wr = w + r * D_ + DE_;
        const float* re = rel + id * DR_;
        float s = 0.f;
        #pragma unroll 8
        for (int d = 0; d < DR_; ++d) {
            float v = re[d];
            s += wr[d] * v * v;
        }
        c[id] = s;
    }
}

// ---------------------------------------------------------------------------
// Main kernel: one WG (8 waves) computes a 64x128 tile of out[b,r,:,:]
//   S = (node_b * w_r) @ node_b^T  + c[b,r],  then sigmoid.
// Each wave: 32x32 subtile via 2x2 grid of v_wmma_f32_16x16x4_f32, K=128.
// ---------------------------------------------------------------------------
__global__ __launch_bounds__(256) void distmult_kernel(
    const float* __restrict__ node,
    const float* __restrict__ w,
    const float* __restrict__ cconst,
    float* __restrict__ out) {

    __shared__ float lA[TILE_M * LDA];   // scaled rows i0..i0+63   (33.8 KB)
    __shared__ float lB[TILE_N * LDA];   // raw rows    j0..j0+127  (67.6 KB)

    const int br = blockIdx.z;           // b*8 + r
    const int b  = br >> 3;
    const int r  = br & 7;
    const int i0 = blockIdx.y * TILE_M;
    const int j0 = blockIdx.x * TILE_N;

    const float* nb = node + (size_t)b * N_ * DE_;
    const float* wr = w + r * D_;
    const int tid = threadIdx.x;

    // ---- fill LDS: A block (scaled by w_r), float4 transfers ----
    // 64 rows * 32 float4/row = 2048 float4 -> 8 per thread
    #pragma unroll
    for (int it = 0; it < (TILE_M * DE_ / 4) / 256; ++it) {
        int idx  = tid + it * 256;
        int row  = idx >> 5;             // 32 float4 per row
        int col4 = idx & 31;
        float4 v = *(const float4*)(nb + (size_t)(i0 + row) * DE_ + col4 * 4);
        float4 s = *(const float4*)(wr + col4 * 4);
        float4 o;
        o.x = v.x * s.x; o.y = v.y * s.y; o.z = v.z * s.z; o.w = v.w * s.w;
        *(float4*)(&lA[row * LDA + col4 * 4]) = o;
    }
    // ---- fill LDS: B block (raw rows, acts as node_b^T source) ----
    // 128 rows * 32 float4/row = 4096 float4 -> 16 per thread
    #pragma unroll
    for (int it = 0; it < (TILE_N * DE_ / 4) / 256; ++it) {
        int idx  = tid + it * 256;
        int row  = idx >> 5;
        int col4 = idx & 31;
        *(float4*)(&lB[row * LDA + col4 * 4]) =
            *(const float4*)(nb + (size_t)(j0 + row) * DE_ + col4 * 4);
    }
    __syncthreads();

    const int lane   = tid & 31;
    const int wave   = tid >> 5;
    const int wm     = (wave & 1) * 32;   // wave's M offset in tile (0/32)
    const int wn     = (wave >> 1) * 32;  // wave's N offset in tile (0/32/64/96)
    const int lane16 = lane & 15;
    const int laneHi = lane >> 4;         // 0: K=0,1 ; 1: K=2,3  (16x4 f32 A layout)

    v8f acc00 = {}, acc01 = {}, acc10 = {}, acc11 = {};

    #pragma unroll 4
    for (int kk = 0; kk < DE_; kk += 4) {
        const int kc = kk + 2 * laneHi;
        v2f a0 = *(const v2f*)(&lA[(wm +      lane16) * LDA + kc]);
        v2f a1 = *(const v2f*)(&lA[(wm + 16 + lane16) * LDA + kc]);
        v2f b0 = *(const v2f*)(&lB[(wn +      lane16) * LDA + kc]);
        v2f b1 = *(const v2f*)(&lB[(wn + 16 + lane16) * LDA + kc]);
        acc00 = __builtin_amdgcn_wmma_f32_16x16x4_f32(false, a0, false, b0, (short)0, acc00, false, false);
        acc01 = __builtin_amdgcn_wmma_f32_16x16x4_f32(false, a0, false, b1, (short)0, acc01, false, false);
        acc10 = __builtin_amdgcn_wmma_f32_16x16x4_f32(false, a1, false, b0, (short)0, acc10, false, false);
        acc11 = __builtin_amdgcn_wmma_f32_16x16x4_f32(false, a1, false, b1, (short)0, acc11, false, false);
    }

    const float cadd = cconst[br];
    float* outbr = out + (size_t)br * N_ * N_;

    // C/D layout: VGPR v holds M = v + 8*laneHi, N = lane16 (within 16x16 block)
    #define STORE_TILE(ACC, MI, NJ)                                           \
        {                                                                     \
            const int col = j0 + wn + (NJ) + lane16;                          \
            _Pragma("unroll")                                                 \
            for (int v = 0; v < 8; ++v) {                                     \
                const int row = i0 + wm + (MI) + v + 8 * laneHi;              \
                float x = (ACC)[v] + cadd;                                    \
                outbr[(size_t)row * N_ + col] = 1.f / (1.f + __expf(-x));     \
            }                                                                 \
        }

    STORE_TILE(acc00,  0,  0)
    STORE_TILE(acc01,  0, 16)
    STORE_TILE(acc10, 16,  0)
    STORE_TILE(acc11, 16, 16)
    #undef STORE_TILE
}

// ---------------------------------------------------------------------------
extern "C" void kernel_launch(void* const* d_in, const int* in_sizes, int n_in,
                              void* d_out, int out_size, void* d_ws, size_t ws_size,
                              hipStream_t stream) {
    const float* node = (const float*)d_in[0];   // [8,1024,128]
    const float* rel  = (const float*)d_in[1];   // [8,8,64]
    const float* w    = (const float*)d_in[2];   // [8,192]
    float* out = (float*)d_out;                  // [8,8,1024,1024]
    float* c   = (float*)d_ws;                   // 64 floats scratch

    relconst_kernel<<<1, 64, 0, stream>>>(rel, w, c);

    dim3 grid(N_ / TILE_N, N_ / TILE_M, B_ * R_);  // 8 x 16 x 64
    distmult_kernel<<<grid, 256, 0, stream>>>(node, w, c, out);
}